// MultiClassNms_1769526526007
// MI455X (gfx1250) — compile-verified
//
#include <hip/hip_runtime.h>
#include <hip/hip_bf16.h>

#define BATCH   8
#define NBOX    1024
#define NCLS    32
#define MAXOUT  300
#define NTH     1024
#define IOU_THR   0.5f
#define SCORE_THR 0.5f

typedef float v2f __attribute__((ext_vector_type(2)));
typedef float v8f __attribute__((ext_vector_type(8)));

#if __has_builtin(__builtin_amdgcn_wmma_f32_16x16x4_f32)
#define HAVE_WMMA_F32X4 1
#endif

// Monotone float -> uint32 order map (handles -inf), and its inverse.
__device__ __forceinline__ unsigned int ford(float x) {
    unsigned int b = __float_as_uint(x);
    return (b & 0x80000000u) ? ~b : (b | 0x80000000u);
}
__device__ __forceinline__ float finvord(unsigned int o) {
    return __uint_as_float((o & 0x80000000u) ? (o ^ 0x80000000u) : ~o);
}

__device__ __forceinline__ unsigned int ballot32(bool p) {
#if __has_builtin(__builtin_amdgcn_ballot_w32)
    return __builtin_amdgcn_ballot_w32(p);
#else
    return (unsigned int)__ballot((int)p);
#endif
}

// Dynamic LDS, shared symbol reused by both kernels (cast per kernel).
extern __shared__ char s_dyn[];

// ---------------------------------------------------------------------------
// Kernel 1: one workgroup per (batch, class).
//  - stable descending sort of scores (bitonic on packed 64-bit keys)
//  - parallel suppression-bitmask build (16x16 tiles, WMMA area-sum + ballot)
//  - sequential bitmask greedy reduction (32 LDS words per step)
//  - compact kept detections (capped at 300) into packed global keys
// ---------------------------------------------------------------------------
__global__ __launch_bounds__(NTH)
void nms_per_class(const float* __restrict__ boxes,
                   const float* __restrict__ scores,
                   unsigned long long* __restrict__ keys_out, // [B][C][MAXOUT]
                   int* __restrict__ cnt_out)                 // [B][C]
{
    unsigned int* sup = (unsigned int*)s_dyn;  // [NBOX][32] suppression bitmask (128 KB)

    __shared__ float4             sb[NBOX];    // sorted boxes (y1,x1,y2,x2)
    __shared__ float              sarea[NBOX]; // areas of sorted boxes
    __shared__ unsigned long long skey[NBOX];  // sort keys
    __shared__ int                soidx[NBOX]; // original index per sorted pos
    __shared__ unsigned int       kept[NBOX / 32];
    __shared__ int                scanb[NTH];

    const int tid = threadIdx.x;
    const int b   = blockIdx.x / NCLS;
    const int c   = blockIdx.x % NCLS;

    // Hint the box rows for this batch into cache early (global_prefetch_b8).
    __builtin_prefetch(boxes + (size_t)(b * NBOX) * 4 + tid * 4, 0, 3);

    // ---- build per-class sort keys: (ord(score) << 32) | ~idx  (desc, stable)
    float s = scores[(size_t)(b * NBOX + tid) * NCLS + c];
    bool  valid = s > SCORE_THR;
    float sv = valid ? s : -INFINITY;
    skey[tid] = ((unsigned long long)ford(sv) << 32) | (unsigned int)(~(unsigned int)tid);
    __syncthreads();

    // ---- bitonic sort, descending
    for (int k = 2; k <= NBOX; k <<= 1) {
        for (int j = k >> 1; j > 0; j >>= 1) {
            int ixj = tid ^ j;
            if (ixj > tid) {
                unsigned long long a = skey[tid], bb = skey[ixj];
                bool desc = ((tid & k) == 0);
                if (desc ? (a < bb) : (a > bb)) { skey[tid] = bb; skey[ixj] = a; }
            }
            __syncthreads();
        }
    }

    // ---- gather sorted boxes, areas, validity; init kept bitmask; clear sup
    unsigned long long myk = skey[tid];
    int oidx = (int)(~(unsigned int)(myk & 0xFFFFFFFFull));
    soidx[tid] = oidx;
    float4 bx = ((const float4*)boxes)[b * NBOX + oidx];
    sb[tid]    = bx;
    sarea[tid] = (bx.z - bx.x) * (bx.w - bx.y);
    float ssc  = finvord((unsigned int)(myk >> 32));
    bool  svalid = ssc > SCORE_THR;                // -inf entries fail
    unsigned int vb = ballot32(svalid);
    if ((tid & 31) == 0) kept[tid >> 5] = vb;
    for (int t = tid; t < NBOX * 32; t += NTH) sup[t] = 0u;
    __syncthreads();

    // ---- phase A: parallel suppression matrix, 16x16 tiles over 32 waves
    const int wave = tid >> 5, lane = tid & 31;
    const int nl = lane & 15;
    const bool lo = lane < 16;
    int tcnt = 0;
    for (int ti = 0; ti < NBOX / 16; ++ti) {
        for (int tj = ti; tj < NBOX / 16; ++tj) {
            if ((tcnt++ & 31) != wave) continue;   // wave-uniform: EXEC all-ones inside

            // asum tile via WMMA: D[m][n] = area_m*1 + 1*area_n  (16x16x4 f32)
            v8f acc = {};
#ifdef HAVE_WMMA_F32X4
            {
                v2f av, bv;
                av.x = lo ? sarea[ti * 16 + nl] : 0.0f;  // A[m][0]=area_m (lanes<16: K=0,1)
                av.y = lo ? 1.0f : 0.0f;                 // A[m][1]=1      (lanes>=16: K=2,3 -> 0)
                bv.x = lo ? 1.0f : 0.0f;                 // B[0][n]=1      (lanes>=16: K=2 -> 0)
                bv.y = lo ? sarea[tj * 16 + nl] : 0.0f;  // B[1][n]=area_n
                acc = __builtin_amdgcn_wmma_f32_16x16x4_f32(
                        false, av, false, bv, (short)0, acc, false, false);
            }
#endif
#pragma unroll
            for (int v = 0; v < 8; ++v) {
                int m = v + (lo ? 0 : 8);
                int i = ti * 16 + m;
                int j = tj * 16 + nl;
                float4 bi = sb[i];
                float4 bj = sb[j];
                float yy1 = fmaxf(bi.x, bj.x);
                float xx1 = fmaxf(bi.y, bj.y);
                float yy2 = fminf(bi.z, bj.z);
                float xx2 = fminf(bi.w, bj.w);
                float inter = fmaxf(yy2 - yy1, 0.0f) * fmaxf(xx2 - xx1, 0.0f);
#ifdef HAVE_WMMA_F32X4
                float asum = acc[v];
#else
                float asum = sarea[i] + sarea[j];
#endif
                // iou > thr  <=>  inter > thr*(union), union > 0 always (sizes >= 1)
                bool p = (j > i) && (inter > IOU_THR * (asum - inter));
                unsigned int bal = ballot32(p);      // bits[15:0]: row ti*16+v, bits[31:16]: row +8
                if (lane == 0) {
                    unsigned int w0 = bal & 0xFFFFu;
                    unsigned int w1 = (bal >> 16) & 0xFFFFu;
                    int word = tj >> 1;
                    int sh   = (tj & 1) * 16;
                    if (w0) atomicOr(&sup[(ti * 16 + v)     * 32 + word], w0 << sh);
                    if (w1) atomicOr(&sup[(ti * 16 + v + 8) * 32 + word], w1 << sh);
                }
            }
        }
    }
    __syncthreads();

    // ---- phase B: sequential greedy reduction over bitmask rows
    for (int i = 0; i < NBOX; ++i) {
        __syncthreads();
        if ((kept[i >> 5] >> (i & 31)) & 1u) {
            if (tid < 32) kept[tid] &= ~sup[i * 32 + tid];  // sup[i] only has bits j > i
        }
    }
    __syncthreads();

    // ---- compact kept detections (ascending sorted position = score desc), cap 300
    bool kp = ((kept[tid >> 5] >> (tid & 31)) & 1u) != 0u;
    scanb[tid] = kp ? 1 : 0;
    __syncthreads();
    for (int off = 1; off < NTH; off <<= 1) {
        int vv = (tid >= off) ? scanb[tid - off] : 0;
        __syncthreads();
        scanb[tid] += vv;
        __syncthreads();
    }
    int rank  = scanb[tid] - (kp ? 1 : 0);
    int total = scanb[NTH - 1];
    if (kp && rank < MAXOUT) {
        unsigned int f = (unsigned int)(c * NBOX + tid);   // class-major flat idx (15 bits)
        unsigned long long key =
              ((unsigned long long)ford(ssc) << 32)
            | ((unsigned long long)((~f) & 0x7FFFu) << 10) // desc key => ascending f on ties
            | (unsigned long long)(soidx[tid] & 0x3FF);    // original box idx (10 bits)
        keys_out[(size_t)(b * NCLS + c) * MAXOUT + rank] = key;
    }
    if (tid == 0) cnt_out[b * NCLS + c] = total < MAXOUT ? total : MAXOUT;
}

// ---------------------------------------------------------------------------
// Kernel 2: one workgroup per batch. Global merge:
//  - gather <= 32*300 candidate keys, bitonic-sort 16K keys descending in LDS
//  - take top-300, dedup by original box id (atomicMin first-occurrence)
//  - emit present boxes in ascending box-id order, zero-fill the rest
// ---------------------------------------------------------------------------
__global__ __launch_bounds__(NTH)
void nms_merge(const float* __restrict__ boxes,
               const unsigned long long* __restrict__ keys, // [B][C][MAXOUT]
               const int* __restrict__ cnt,                 // [B][C]
               float* __restrict__ out)
{
    unsigned long long* lk = (unsigned long long*)s_dyn;    // 16384 keys (128 KB)

    __shared__ int first[NBOX];
    __shared__ int scanb[NTH];

    const int tid = threadIdx.x;
    const int b   = blockIdx.x;
    const int TOT   = NCLS * MAXOUT;  // 9600
    const int SORTN = 16384;          // next pow2

    for (int t = tid; t < SORTN; t += NTH) {
        unsigned long long k = 0ull;                        // pad: sorts last (scores > 0.5 => key hi >= 0xBF000000)
        if (t < TOT) {
            int cc = t / MAXOUT;
            int e  = t - cc * MAXOUT;
            if (e < cnt[b * NCLS + cc])
                k = keys[(size_t)(b * NCLS + cc) * MAXOUT + e];
        }
        lk[t] = k;
    }
    __syncthreads();

    // bitonic sort descending, 16 elements per thread per stage
    for (int k = 2; k <= SORTN; k <<= 1) {
        for (int j = k >> 1; j > 0; j >>= 1) {
            for (int t = tid; t < SORTN; t += NTH) {
                int ixj = t ^ j;
                if (ixj > t) {
                    unsigned long long a = lk[t], bb = lk[ixj];
                    bool desc = ((t & k) == 0);
                    if (desc ? (a < bb) : (a > bb)) { lk[t] = bb; lk[ixj] = a; }
                }
            }
            __syncthreads();
        }
    }

    // dedup within top-300 by original box id: first (= highest score) occurrence
    first[tid] = MAXOUT;                                    // NTH == NBOX
    __syncthreads();
    if (tid < MAXOUT) {
        unsigned long long kk = lk[tid];
        if (kk != 0ull) atomicMin(&first[(int)(kk & 0x3FFull)], tid);
    }
    __syncthreads();

    bool pres = first[tid] < MAXOUT;
    scanb[tid] = pres ? 1 : 0;
    __syncthreads();
    for (int off = 1; off < NTH; off <<= 1) {
        int vv = (tid >= off) ? scanb[tid - off] : 0;
        __syncthreads();
        scanb[tid] += vv;
        __syncthreads();
    }
    int P = scanb[NTH - 1];                                 // <= 300 distinct boxes

    float* ob  = out;                                       // [B][300][4]
    float* osf = out + (size_t)BATCH * MAXOUT * 4;          // [B][300]
    float* ocl = osf + (size_t)BATCH * MAXOUT;              // [B][300]

    if (pres) {                                             // tid = original box id, ascending order
        int r = scanb[tid] - 1;
        unsigned long long kk = lk[first[tid]];
        float sc = finvord((unsigned int)(kk >> 32));
        int   f  = (int)((~(kk >> 10)) & 0x7FFFull);
        int   cls = f >> 10;
        float4 bx = ((const float4*)boxes)[b * NBOX + tid];
        ((float4*)ob)[b * MAXOUT + r] = bx;
        osf[b * MAXOUT + r] = sc;
        ocl[b * MAXOUT + r] = (float)cls;
    }
    if (tid < MAXOUT && tid >= P) {
        float4 z = make_float4(0.f, 0.f, 0.f, 0.f);
        ((float4*)ob)[b * MAXOUT + tid] = z;
        osf[b * MAXOUT + tid] = 0.f;
        ocl[b * MAXOUT + tid] = 0.f;
    }
}

// ---------------------------------------------------------------------------
extern "C" void kernel_launch(void* const* d_in, const int* in_sizes, int n_in,
                              void* d_out, int out_size, void* d_ws, size_t ws_size,
                              hipStream_t stream) {
    (void)in_sizes; (void)n_in; (void)out_size; (void)ws_size;
    const float* boxes  = (const float*)d_in[0];   // [8,1024,4]
    const float* scores = (const float*)d_in[1];   // [8,1024,32]

    unsigned long long* keys = (unsigned long long*)d_ws;            // 614400 B
    int* cnt = (int*)((char*)d_ws + (size_t)BATCH * NCLS * MAXOUT * sizeof(unsigned long long));

    const size_t dyn1 = (size_t)NBOX * 32 * sizeof(unsigned int);        // 128 KB sup bitmask
    const size_t dyn2 = (size_t)16384 * sizeof(unsigned long long);      // 128 KB sort keys

    nms_per_class<<<BATCH * NCLS, NTH, dyn1, stream>>>(boxes, scores, keys, cnt);
    nms_merge<<<BATCH, NTH, dyn2, stream>>>(boxes, keys, cnt, (float*)d_out);
}